// ScanExample_55903294324887
// MI455X (gfx1250) — compile-verified
//
#include <hip/hip_runtime.h>
#include <hip/hip_bf16.h>

// Output = init + column-sum(xs) over 65536 rows x 512 cols (f32).
// Pass 1: per (col-tile, row-slab) partial column sums via V_WMMA_F32_16X16X4_F32
//         (A = ones 16x4, B = 4x16 tile of xs -> C accumulates column sums).
// Pass 2: deterministic fixed-order reduction of slab partials + init.

typedef __attribute__((ext_vector_type(2))) float v2f;
typedef __attribute__((ext_vector_type(8))) float v8f;

#define NCOLS 512

__global__ __launch_bounds__(256)
void ScanColSum_partial(const float* __restrict__ xs,
                        float* __restrict__ ws,
                        int rowsPerSlab) {
    const int lane  = threadIdx.x & 31;     // wave32
    const int wave  = threadIdx.x >> 5;     // 8 waves per block
    const int colBase = (blockIdx.x * 8 + wave) * 16;   // 16-col tile per wave
    const int col     = colBase + (lane & 15);
    const int half    = lane >> 4;          // 0: lanes 0-15, 1: lanes 16-31
    const int rowStart = blockIdx.y * rowsPerSlab;

    // B layout (4x16 f32): VGPR0 = {K0 | K2}, VGPR1 = {K1 | K3} across half-waves.
    // With A == ones, only the N(=lane) mapping matters for the column sum.
    const float* p = xs + (size_t)(rowStart + 2 * half) * NCOLS + col;

    v2f a; a.x = 1.0f; a.y = 1.0f;          // A = ones (16x4)
    v8f c0 = {}; v8f c1 = {};               // two accumulators for WMMA ILP

    for (int r = 0; r < rowsPerSlab; r += 8) {
        __builtin_prefetch((const void*)(p + 16 * NCOLS), 0, 1); // 2 iters ahead
        v2f b0, b1;
        b0.x = p[0];            // rows R+2h
        b0.y = p[NCOLS];        // rows R+2h+1
        b1.x = p[4 * NCOLS];        // rows R+4+2h
        b1.y = p[5 * NCOLS];        // rows R+4+2h+1
        c0 = __builtin_amdgcn_wmma_f32_16x16x4_f32(false, a, false, b0,
                                                   (short)0, c0, false, false);
        c1 = __builtin_amdgcn_wmma_f32_16x16x4_f32(false, a, false, b1,
                                                   (short)0, c1, false, false);
        p += 8 * NCOLS;
    }

    // Every row of C holds the same column sums; VGPR0 lanes 0-15 -> N=0..15.
    float s = c0[0] + c1[0];
    if (lane < 16) {
        ws[(size_t)blockIdx.y * NCOLS + col] = s;
    }
}

__global__ __launch_bounds__(256)
void ScanColSum_final(const float* __restrict__ init,
                      const float* __restrict__ ws,
                      float* __restrict__ out,
                      int slabs) {
    const int col = blockIdx.x * blockDim.x + threadIdx.x;
    if (col < NCOLS) {
        float s = init[col];
        // Fixed serial order -> bitwise deterministic across replays.
        for (int i = 0; i < slabs; ++i) {
            s += ws[(size_t)i * NCOLS + col];
        }
        out[col] = s;
    }
}

extern "C" void kernel_launch(void* const* d_in, const int* in_sizes, int n_in,
                              void* d_out, int out_size, void* d_ws, size_t ws_size,
                              hipStream_t stream) {
    const float* init = (const float*)d_in[0];   // (512,)
    const float* xs   = (const float*)d_in[1];   // (65536, 512)
    float* out = (float*)d_out;                  // (512,)
    float* ws  = (float*)d_ws;                   // slab partials

    const int totalRows = 65536;

    // Choose slab count: power of two, <= 256, fitting in ws_size.
    int maxSlabs = (int)(ws_size / (NCOLS * sizeof(float)));
    int slabs = 1;
    while (slabs * 2 <= maxSlabs && slabs * 2 <= 256) slabs *= 2;
    int rowsPerSlab = totalRows / slabs;  // multiple of 8 (slabs is pow2 <= 256)

    dim3 grid1(4, slabs);   // 4 col-groups of 128 cols (8 waves x 16 cols)
    ScanColSum_partial<<<grid1, 256, 0, stream>>>(xs, ws, rowsPerSlab);
    ScanColSum_final<<<2, 256, 0, stream>>>(init, ws, out, slabs);
}